// PrototypicalRefinement_2568390443014
// MI455X (gfx1250) — compile-verified
//
#include <hip/hip_runtime.h>

// ---------------------------------------------------------------------------
// PrototypicalRefinement on MI455X (gfx1250), f32 WMMA path.
//   - hq GEMM (4096x512x1024): 16x64 tile per wave, 4 independent accumulators
//   - small M=32 GEMMs: split-K into partial buffers + deterministic reduce
// ---------------------------------------------------------------------------
static constexpr int Dm      = 1024;   // feature dim
static constexpr int Hh      = 512;    // hidden dim
static constexpr int NWAY    = 20;     // prototypes
static constexpr int NQ      = 4096;   // queries
static constexpr int NITERS  = 3;
static constexpr int MPAD    = 32;     // prototypes padded to 2 WMMA tiles
static constexpr float LN_EPSF   = 1e-5f;
static constexpr float NORM_EPSF = 1e-12f;

typedef __attribute__((ext_vector_type(2))) float v2f;
typedef __attribute__((ext_vector_type(8))) float v8f;

__device__ __forceinline__ v8f wmma4(v2f a, v2f b, v8f c)
{
    return __builtin_amdgcn_wmma_f32_16x16x4_f32(false, a, false, b,
                                                 (short)0, c, false, false);
}

// ---------------------------------------------------------------------------
// Big GEMM: C[M x N] = A[M x K] * B^T, B row-major N x K.
// One wave computes a 16x64 tile: A fragment reused across 4 N-tiles,
// 4 independent accumulators -> back-to-back WMMA issue.
// ---------------------------------------------------------------------------
__global__ __launch_bounds__(256)
void wmma_gemm_f32_nt4(const float* __restrict__ A, int lda,
                       const float* __restrict__ B, int ldb,
                       float* __restrict__ C, int ldc,
                       int M, int N, int K)
{
    const int lane = threadIdx.x & 31;
    const int wv   = threadIdx.x >> 5;
    const int ntn  = N >> 6;                      // 64-wide N tiles
    const int tiles = (M >> 4) * ntn;
    const int t = blockIdx.x * (blockDim.x >> 5) + wv;
    if (t >= tiles) return;                       // uniform per-wave
    const int m0 = (t / ntn) << 4;
    const int n0 = (t % ntn) << 6;

    const int l15 = lane & 15;
    const int hi  = lane >> 4;                    // 0: K={0,1}, 1: K={2,3}
    const int kof = hi << 1;

    const float* Ap  = A + (size_t)(m0 + l15) * lda + kof;
    const float* Bp0 = B + (size_t)(n0 + l15) * ldb + kof;
    const float* Bp1 = Bp0 + (size_t)16 * ldb;
    const float* Bp2 = Bp0 + (size_t)32 * ldb;
    const float* Bp3 = Bp0 + (size_t)48 * ldb;

    v8f c0 = {}, c1 = {}, c2 = {}, c3 = {};
    #pragma unroll 2
    for (int k = 0; k < K; k += 4) {
        v2f a, b0, b1, b2, b3;
        a.x  = Ap[k];  a.y  = Ap[k + 1];
        b0.x = Bp0[k]; b0.y = Bp0[k + 1];
        b1.x = Bp1[k]; b1.y = Bp1[k + 1];
        b2.x = Bp2[k]; b2.y = Bp2[k + 1];
        b3.x = Bp3[k]; b3.y = Bp3[k + 1];
        c0 = wmma4(a, b0, c0);
        c1 = wmma4(a, b1, c1);
        c2 = wmma4(a, b2, c2);
        c3 = wmma4(a, b3, c3);
    }

    const int mrow = m0 + (hi << 3);
    #pragma unroll
    for (int r = 0; r < 8; ++r) {
        float* crow = C + (size_t)(mrow + r) * ldc + l15;
        crow[n0]      = c0[r];
        crow[n0 + 16] = c1[r];
        crow[n0 + 32] = c2[r];
        crow[n0 + 48] = c3[r];
    }
}

// ---------------------------------------------------------------------------
// Split-K GEMM for the small M=32 GEMMs. grid.y = #splits; each split writes a
// private partial C (no atomics -> deterministic). BT=1: B is N x K row-major.
// BT=0: B is K x N row-major.
// ---------------------------------------------------------------------------
template <int BT>
__global__ __launch_bounds__(256)
void wmma_gemm_f32_splitk(const float* __restrict__ A, int lda,
                          const float* __restrict__ B, int ldb,
                          float* __restrict__ Cpart,   // [splits][M*N]
                          int M, int N, int K)
{
    const int lane = threadIdx.x & 31;
    const int wv   = threadIdx.x >> 5;
    const int ntn  = N >> 4;
    const int tiles = (M >> 4) * ntn;
    const int t = blockIdx.x * (blockDim.x >> 5) + wv;
    if (t >= tiles) return;
    const int m0 = (t / ntn) << 4;
    const int n0 = (t % ntn) << 4;

    const int kchunk = K / gridDim.y;             // multiple of 4 by launch
    const int kbeg = blockIdx.y * kchunk;
    const int kend = kbeg + kchunk;

    const int l15 = lane & 15;
    const int hi  = lane >> 4;
    const int kof = hi << 1;

    const float* Ap = A + (size_t)(m0 + l15) * lda + kof;
    v8f c = {};

    if (BT) {
        const float* Bp = B + (size_t)(n0 + l15) * ldb + kof;
        #pragma unroll 4
        for (int k = kbeg; k < kend; k += 4) {
            v2f a, b;
            a.x = Ap[k]; a.y = Ap[k + 1];
            b.x = Bp[k]; b.y = Bp[k + 1];
            c = wmma4(a, b, c);
        }
    } else {
        const float* Bp = B + (size_t)kof * ldb + n0 + l15;
        #pragma unroll 4
        for (int k = kbeg; k < kend; k += 4) {
            v2f a, b;
            a.x = Ap[k]; a.y = Ap[k + 1];
            b.x = Bp[(size_t)k * ldb];
            b.y = Bp[(size_t)(k + 1) * ldb];
            c = wmma4(a, b, c);
        }
    }

    float* Cp = Cpart + (size_t)blockIdx.y * M * N;
    const int mrow = m0 + (hi << 3);
    const int ncol = n0 + l15;
    #pragma unroll
    for (int r = 0; r < 8; ++r)
        Cp[(size_t)(mrow + r) * N + ncol] = c[r];
}

// ---------------------------------------------------------------------------
// out[i] = sum_p parts[p][i] + bias[i % N]   (fixed order -> deterministic)
// ---------------------------------------------------------------------------
__global__ __launch_bounds__(256)
void reduce_parts_kernel(const float* __restrict__ parts, int nparts,
                         int MN, int N, const float* __restrict__ bias,
                         float* __restrict__ out)
{
    const int i = blockIdx.x * blockDim.x + threadIdx.x;
    if (i >= MN) return;
    float s = bias ? bias[i % N] : 0.0f;
    for (int p = 0; p < nparts; ++p)
        s += parts[(size_t)p * MN + i];
    out[i] = s;
}

// ---------------------------------------------------------------------------
// Block-wide sum for blockDim == 256 (8 waves of 32).
// ---------------------------------------------------------------------------
__device__ __forceinline__ float block_sum(float v, float* buf)
{
    #pragma unroll
    for (int off = 16; off; off >>= 1) v += __shfl_xor(v, off, 32);
    const int lane = threadIdx.x & 31;
    const int wv   = threadIdx.x >> 5;
    __syncthreads();
    if (lane == 0) buf[wv] = v;
    __syncthreads();
    float r = buf[0];
    #pragma unroll
    for (int i = 1; i < 8; ++i) r += buf[i];
    return r;
}

// ---------------------------------------------------------------------------
// att[w][q] = sigmoid( sum_h relu(hp[w][h] + hq[q][h] + ab1[h]) * aW2[h] + ab2 )
//             * (1/NQ)
// ---------------------------------------------------------------------------
__global__ __launch_bounds__(256)
void attention_kernel(const float* __restrict__ hp, const float* __restrict__ hq,
                      const float* __restrict__ ab1, const float* __restrict__ aW2,
                      const float* __restrict__ ab2, float* __restrict__ att,
                      float inv_nq)
{
    const int lane = threadIdx.x & 31;
    const int wv   = threadIdx.x >> 5;
    const int q = blockIdx.x * (blockDim.x >> 5) + wv;
    const int w = blockIdx.y;

    const float* hprow = hp + (size_t)w * Hh;
    const float* hqrow = hq + (size_t)q * Hh;

    float acc = 0.0f;
    #pragma unroll 4
    for (int i = lane; i < Hh; i += 32) {
        float v = hprow[i] + hqrow[i] + ab1[i];
        v = v > 0.0f ? v : 0.0f;
        acc = fmaf(v, aW2[i], acc);
    }
    #pragma unroll
    for (int off = 16; off; off >>= 1) acc += __shfl_xor(acc, off, 32);

    if (lane == 0) {
        float z = acc + ab2[0];
        float s = 1.0f / (1.0f + expf(-z));
        att[(size_t)w * NQ + q] = s * inv_nq;
    }
}

// ---------------------------------------------------------------------------
// LayerNorm over D then ReLU. One block (256 thr) per row; rows >= NWAY -> 0.
// ---------------------------------------------------------------------------
__global__ __launch_bounds__(256)
void ln_relu_kernel(const float* __restrict__ x, const float* __restrict__ g,
                    const float* __restrict__ bta, float* __restrict__ xr)
{
    __shared__ float buf[8];
    const int w = blockIdx.x;
    float* outrow = xr + (size_t)w * Dm;
    if (w >= NWAY) {
        for (int i = threadIdx.x; i < Dm; i += 256) outrow[i] = 0.0f;
        return;
    }
    const float* row = x + (size_t)w * Dm;
    float v[Dm / 256];
    float s = 0.0f, s2 = 0.0f;
    #pragma unroll
    for (int j = 0; j < Dm / 256; ++j) {
        v[j] = row[threadIdx.x + j * 256];
        s  += v[j];
        s2 = fmaf(v[j], v[j], s2);
    }
    s  = block_sum(s,  buf);
    s2 = block_sum(s2, buf);
    const float mu   = s  * (1.0f / Dm);
    const float var  = s2 * (1.0f / Dm) - mu * mu;
    const float rstd = rsqrtf(var + LN_EPSF);
    #pragma unroll
    for (int j = 0; j < Dm / 256; ++j) {
        int i = threadIdx.x + j * 256;
        float y = (v[j] - mu) * rstd * g[i] + bta[i];
        outrow[i] = y > 0.0f ? y : 0.0f;
    }
}

// ---------------------------------------------------------------------------
// P <- normalize(P + step * delta). One block per prototype row (0..19).
// ---------------------------------------------------------------------------
__global__ __launch_bounds__(256)
void update_p_kernel(float* __restrict__ Ppad, const float* __restrict__ delta,
                     const float* __restrict__ steps, int it)
{
    __shared__ float buf[8];
    const int w = blockIdx.x;
    const float st = steps[it];
    float* row = Ppad + (size_t)w * Dm;
    const float* drow = delta + (size_t)w * Dm;
    float v[Dm / 256];
    float s2 = 0.0f;
    #pragma unroll
    for (int j = 0; j < Dm / 256; ++j) {
        int i = threadIdx.x + j * 256;
        v[j] = fmaf(st, drow[i], row[i]);
        s2 = fmaf(v[j], v[j], s2);
    }
    s2 = block_sum(s2, buf);
    const float inv = 1.0f / fmaxf(sqrtf(s2), NORM_EPSF);
    #pragma unroll
    for (int j = 0; j < Dm / 256; ++j)
        row[threadIdx.x + j * 256] = v[j] * inv;
}

// ---------------------------------------------------------------------------
// Init: Ppad rows 0..19 = prototypes, rows 20..31 = 0; att rows 20..31 = 0.
// ---------------------------------------------------------------------------
__global__ void init_kernel(const float* __restrict__ proto,
                            float* __restrict__ Ppad, float* __restrict__ att)
{
    const int i = blockIdx.x * blockDim.x + threadIdx.x;   // 0 .. 49151
    if (i < MPAD * Dm) Ppad[i] = (i < NWAY * Dm) ? proto[i] : 0.0f;
    if (i < (MPAD - NWAY) * NQ) att[(size_t)NWAY * NQ + i] = 0.0f;
}

__global__ void copy_out_kernel(const float* __restrict__ src, float* __restrict__ dst)
{
    const int i = blockIdx.x * blockDim.x + threadIdx.x;
    dst[i] = src[i];
}

// ---------------------------------------------------------------------------
extern "C" void kernel_launch(void* const* d_in, const int* in_sizes, int n_in,
                              void* d_out, int out_size, void* d_ws, size_t ws_size,
                              hipStream_t stream)
{
    (void)in_sizes; (void)n_in; (void)out_size; (void)ws_size;

    const float* proto = (const float*)d_in[0];   // (20, 1024)
    const float* Q     = (const float*)d_in[1];   // (4096, 1024)
    const float* aW1   = (const float*)d_in[2];   // (512, 2048)
    const float* ab1   = (const float*)d_in[3];   // (512,)
    const float* aW2   = (const float*)d_in[4];   // (1, 512)
    const float* ab2   = (const float*)d_in[5];   // (1,)
    const float* rW1   = (const float*)d_in[6];   // (1024, 2048)
    const float* rb1   = (const float*)d_in[7];   // (1024,)
    const float* ln_g  = (const float*)d_in[8];   // (1024,)
    const float* ln_b  = (const float*)d_in[9];   // (1024,)
    const float* rW2   = (const float*)d_in[10];  // (1024, 1024)
    const float* rb2   = (const float*)d_in[11];  // (1024,)
    const float* steps = (const float*)d_in[12];  // (3,)

    float* ws    = (float*)d_ws;
    float* hq    = ws;                              // 4096*512
    float* Ppad  = hq    + (size_t)NQ * Hh;         // 32*1024
    float* hp    = Ppad  + (size_t)MPAD * Dm;       // 32*512
    float* att   = hp    + (size_t)MPAD * Hh;       // 32*4096
    float* agg   = att   + (size_t)MPAD * NQ;       // 32*1024
    float* x     = agg   + (size_t)MPAD * Dm;       // 32*1024
    float* xr    = x     + (size_t)MPAD * Dm;       // 32*1024
    float* delta = xr    + (size_t)MPAD * Dm;       // 32*1024
    float* parts = delta + (size_t)MPAD * Dm;       // 8*32*1024 partials

    init_kernel<<<192, 256, 0, stream>>>(proto, Ppad, att);

    // hq = Q @ aW1[:, D:].T   (M=4096, N=512, K=1024) -- 16x64 tiles
    wmma_gemm_f32_nt4<<<(NQ / 16) * (Hh / 64) / 8, 256, 0, stream>>>(
        Q, Dm, aW1 + Dm, 2 * Dm, hq, Hh, NQ, Hh, Dm);

    for (int it = 0; it < NITERS; ++it) {
        // hp = P @ aW1[:, :D].T : split-K 4 -> reduce  (32 x 512)
        wmma_gemm_f32_splitk<1><<<dim3(8, 4), 256, 0, stream>>>(
            Ppad, Dm, aW1, 2 * Dm, parts, MPAD, Hh, Dm);
        reduce_parts_kernel<<<(MPAD * Hh) / 256, 256, 0, stream>>>(
            parts, 4, MPAD * Hh, Hh, nullptr, hp);

        // att = sigmoid(relu(hp + hq + ab1) . aW2 + ab2) / NQ
        dim3 agrid(NQ / 8, NWAY);
        attention_kernel<<<agrid, 256, 0, stream>>>(hp, hq, ab1, aW2, ab2, att,
                                                    1.0f / (float)NQ);

        // agg = att @ Q : split-K 8 -> reduce  (32 x 1024, K=4096)
        wmma_gemm_f32_splitk<0><<<dim3(16, 8), 256, 0, stream>>>(
            att, NQ, Q, Dm, parts, MPAD, Dm, NQ);
        reduce_parts_kernel<<<(MPAD * Dm) / 256, 256, 0, stream>>>(
            parts, 8, MPAD * Dm, Dm, nullptr, agg);

        // x = [P, agg] @ rW1.T + rb1 : two split-K GEMMs into disjoint
        // partial slices, reduced together with the bias (concat fused away)
        wmma_gemm_f32_splitk<1><<<dim3(16, 4), 256, 0, stream>>>(
            Ppad, Dm, rW1, 2 * Dm, parts, MPAD, Dm, Dm);
        wmma_gemm_f32_splitk<1><<<dim3(16, 4), 256, 0, stream>>>(
            agg, Dm, rW1 + Dm, 2 * Dm, parts + (size_t)4 * MPAD * Dm,
            MPAD, Dm, Dm);
        reduce_parts_kernel<<<(MPAD * Dm) / 256, 256, 0, stream>>>(
            parts, 8, MPAD * Dm, Dm, rb1, x);

        // xr = relu(LN(x) * g + b)
        ln_relu_kernel<<<MPAD, 256, 0, stream>>>(x, ln_g, ln_b, xr);

        // delta = xr @ rW2.T + rb2 : split-K 4 -> reduce
        wmma_gemm_f32_splitk<1><<<dim3(16, 4), 256, 0, stream>>>(
            xr, Dm, rW2, Dm, parts, MPAD, Dm, Dm);
        reduce_parts_kernel<<<(MPAD * Dm) / 256, 256, 0, stream>>>(
            parts, 4, MPAD * Dm, Dm, rb2, delta);

        // P = normalize(P + step * delta)
        update_p_kernel<<<NWAY, 256, 0, stream>>>(Ppad, delta, steps, it);
    }

    copy_out_kernel<<<(NWAY * Dm) / 256, 256, 0, stream>>>(Ppad, (float*)d_out);
}